// CapLayer_33517924778159
// MI455X (gfx1250) — compile-verified
//
#include <hip/hip_runtime.h>

// Capsule routing, fully fused for MI455X (gfx1250, wave32).
// Shapes: x[64, 32*8, 14,14], W[32,160,8], bias[32,160] -> v[64,10,16]
#define BS   64
#define NG   32
#define IND  8
#define NJ   10
#define ND   16
#define JD   160           // NJ*ND
#define NP   196           // 14*14
#define XS_STRIDE 257      // padded to dodge half-wave bank aliasing
#define WS_STRIDE 9
#define PT_STRIDE 65       // stride-65 so row-varying column reads hit distinct banks

typedef __attribute__((ext_vector_type(2))) float v2f;
typedef __attribute__((ext_vector_type(8))) float v8f;

__global__ __launch_bounds__(256) void zero_kernel(float* p, int n) {
  int i = blockIdx.x * 256 + threadIdx.x;
  if (i < n) p[i] = 0.0f;
}

// One workgroup per (b, g).  2 waves.  Computes pred = W_g @ X_bg + bias via
// V_WMMA_F32_16X16X4_F32 into LDS in 64-column blocks, then runs one routing
// iteration (logits from vsum, softmax over J, weighted reduction) and writes
// the per-group partial S into Spart[(b*NG+g)*JD + o].  No atomics.
__global__ __launch_bounds__(64) void route_kernel(
    const float* __restrict__ x, const float* __restrict__ W,
    const float* __restrict__ bias, const float* __restrict__ vsum,
    float* __restrict__ Spart) {
  __shared__ float Xs[IND][XS_STRIDE];      // input tile, zero-padded cols 196..255
  __shared__ float Ws[JD][WS_STRIDE];       // W_g, padded stride
  __shared__ float pred[JD][PT_STRIDE];     // 160 x 64 pred block
  __shared__ float ctile[NJ][PT_STRIDE];    // softmax coefficients per column
  __shared__ float vs[JD];                  // running v-sum for this batch
  __shared__ float bsh[JD];                 // bias

  const int tid  = threadIdx.x;
  const int b    = blockIdx.x / NG;
  const int g    = blockIdx.x % NG;
  const int lane = tid & 31;
  const int wave = tid >> 5;

  // cooperative loads -------------------------------------------------------
  for (int idx = tid; idx < IND * NP; idx += 64) {
    int k = idx / NP, p = idx - k * NP;
    Xs[k][p] = x[(b * (NG * IND) + g * IND + k) * NP + p];
  }
  for (int idx = tid; idx < IND * (256 - NP); idx += 64) {   // zero pad
    int k = idx / (256 - NP), c = NP + (idx - k * (256 - NP));
    Xs[k][c] = 0.0f;
  }
  for (int idx = tid; idx < JD * IND; idx += 64) {
    int o = idx >> 3, k = idx & 7;
    Ws[o][k] = W[(g * JD + o) * IND + k];
  }
  for (int idx = tid; idx < JD; idx += 64) {
    bsh[idx] = bias[g * JD + idx];
    vs[idx]  = vsum[b * JD + idx];
  }

  // per-thread row ownership for the reduction (rows r0,r1[,r2]) ------------
  float acc0 = 0.0f, acc1 = 0.0f, acc2 = 0.0f;
  const int r0 = tid, r1 = tid + 64, r2 = tid + 128;
  const int j0 = r0 >> 4, j1 = r1 >> 4, j2 = r2 >> 4;

  for (int nb = 0; nb < 4; ++nb) {
    const int base  = nb * 64;
    const int width = (nb == 3) ? (NP - 192) : 64;
    __syncthreads();

    // ---- phase 1: WMMA pred block (10 M-tiles x 4 N16-tiles, K=8 = 2xK4) --
    for (int t = wave; t < 40; t += 2) {
      const int mt = t % 10;
      const int nt = t / 10;
      const int mrow  = mt * 16 + (lane & 15);
      const int khalf = (lane >> 4) * 2;                // A/B 16x16x4 fp32 layout
      const int bcol  = base + nt * 16 + (lane & 15);
      v8f c = {};
      v2f a, bb;
      a.x  = Ws[mrow][khalf + 0];
      a.y  = Ws[mrow][khalf + 1];
      bb.x = Xs[khalf + 0][bcol];
      bb.y = Xs[khalf + 1][bcol];
      c = __builtin_amdgcn_wmma_f32_16x16x4_f32(false, a, false, bb, (short)0, c,
                                                false, false);
      a.x  = Ws[mrow][4 + khalf + 0];
      a.y  = Ws[mrow][4 + khalf + 1];
      bb.x = Xs[4 + khalf + 0][bcol];
      bb.y = Xs[4 + khalf + 1][bcol];
      c = __builtin_amdgcn_wmma_f32_16x16x4_f32(false, a, false, bb, (short)0, c,
                                                false, false);
      const int colL  = nt * 16 + (lane & 15);
      const int rbase = mt * 16 + (lane >> 4) * 8;      // C/D 8-VGPR layout
#pragma unroll
      for (int r = 0; r < 8; ++r)
        pred[rbase + r][colL] = c[r] + bsh[rbase + r];
    }
    __syncthreads();

    // ---- phase 2a: logits + softmax over J per column ---------------------
    if (tid < width) {
      float bj[NJ];
#pragma unroll
      for (int j = 0; j < NJ; ++j) {
        float acc = 0.0f;
#pragma unroll
        for (int d = 0; d < ND; ++d)
          acc += vs[j * 16 + d] * pred[j * 16 + d][tid];
        bj[j] = acc;
      }
      float m = bj[0];
#pragma unroll
      for (int j = 1; j < NJ; ++j) m = fmaxf(m, bj[j]);
      float ssum = 0.0f;
#pragma unroll
      for (int j = 0; j < NJ; ++j) { bj[j] = expf(bj[j] - m); ssum += bj[j]; }
      const float inv = 1.0f / ssum;
#pragma unroll
      for (int j = 0; j < NJ; ++j) ctile[j][tid] = bj[j] * inv;
    }
    __syncthreads();

    // ---- phase 2b: S[o] += sum_col c[j(o)][col] * pred[o][col] ------------
    for (int col = 0; col < width; ++col) {
      acc0 += ctile[j0][col] * pred[r0][col];
      acc1 += ctile[j1][col] * pred[r1][col];
      if (tid < 32) acc2 += ctile[j2][col] * pred[r2][col];
    }
  }

  float* outp = Spart + (b * NG + g) * JD;
  outp[r0] = acc0;
  outp[r1] = acc1;
  if (tid < 32) outp[r2] = acc2;
}

// Reduce partials over groups, squash, and either update the running vsum
// (iters 0,1) or emit the final v (iter 2).
__global__ __launch_bounds__(JD) void finalize_kernel(
    const float* __restrict__ Spart, float* __restrict__ vsum,
    float* __restrict__ out, int is_last) {
  __shared__ float sq[JD];
  const int b = blockIdx.x;
  const int o = threadIdx.x;
  float s = 0.0f;
  for (int g = 0; g < NG; ++g) s += Spart[(b * NG + g) * JD + o];
  sq[o] = s * s;
  __syncthreads();
  const int j16 = o & ~15;
  float nsq = 0.0f;
#pragma unroll
  for (int d = 0; d < 16; ++d) nsq += sq[j16 + d];
  const float nrm   = sqrtf(nsq);
  const float coeff = (nsq / (1.0f + nsq)) / (nrm + 1e-20f);
  const float v = s * coeff;
  if (is_last) out[b * JD + o] = v;
  else         vsum[b * JD + o] += v;
}

extern "C" void kernel_launch(void* const* d_in, const int* in_sizes, int n_in,
                              void* d_out, int out_size, void* d_ws, size_t ws_size,
                              hipStream_t stream) {
  const float* x    = (const float*)d_in[0];
  const float* W    = (const float*)d_in[1];
  const float* bias = (const float*)d_in[2];
  float* out  = (float*)d_out;
  float* ws   = (float*)d_ws;
  float* vsum  = ws;               // BS*JD floats (40 KB)
  float* Spart = ws + BS * JD;     // BS*NG*JD floats (1.31 MB)

  zero_kernel<<<(BS * JD + 255) / 256, 256, 0, stream>>>(vsum, BS * JD);
  for (int it = 0; it < 3; ++it) {
    route_kernel<<<BS * NG, 64, 0, stream>>>(x, W, bias, vsum, Spart);
    finalize_kernel<<<BS, JD, 0, stream>>>(Spart, vsum, out, it == 2 ? 1 : 0);
  }
}